// CoordinateRefiner_75222057222748
// MI455X (gfx1250) — compile-verified
//
#include <hip/hip_runtime.h>
#include <hip/hip_bf16.h>
#include <math.h>

// Problem constants (match reference)
#define BQ 4
#define LQ 512
#define D0Q 640
#define PQ 128
#define CQ 32
#define NQ (BQ*LQ)      // 2048 nodes
#define RADP 132        // RAD_IN = 130 padded to multiple of 4 (WMMA K-step)

typedef __attribute__((ext_vector_type(2))) float v2f;
typedef __attribute__((ext_vector_type(8))) float v8f;

__device__ __forceinline__ float wred32(float v){
  v += __shfl_xor(v, 1, 32);
  v += __shfl_xor(v, 2, 32);
  v += __shfl_xor(v, 4, 32);
  v += __shfl_xor(v, 8, 32);
  v += __shfl_xor(v, 16, 32);
  return v;
}
__device__ __forceinline__ float wred8(float v){
  v += __shfl_xor(v, 1, 32);
  v += __shfl_xor(v, 2, 32);
  v += __shfl_xor(v, 4, 32);
  return v;
}
__device__ __forceinline__ void atomicMaxFloat(float* addr, float val){
  // monotonic int mapping trick; buffer initialized to -INF (0xFF800000)
  if (val >= 0.f) atomicMax((int*)addr, __float_as_int(val));
  else            atomicMin((unsigned int*)addr, (unsigned int)__float_as_int(val));
}

// ---------------------------------------------------------------------------
// Generic fp32 WMMA GEMM: C[M, NCT*16] (+)= opt_relu(A[M,K] @ W[K,NCT*16] + bias)
// One wave owns a 16-row tile and all NCT column tiles; K consumed 4 at a time
// with V_WMMA_F32_16X16X4_F32.
// A operand layout (ISA 7.12.2, 32-bit A 16x4): lanes 0-15 -> K=0,1 of row=lane;
// lanes 16-31 -> K=2,3 of row=lane-16.  B mirrors with N=lane.
// ---------------------------------------------------------------------------
template<int NCT, bool RELU, bool ADD>
__global__ void wmma_gemm(const float* __restrict__ A, const float* __restrict__ W,
                          const float* __restrict__ bias, float* __restrict__ C,
                          int M, int K, int lda)
{
  const int ldw = NCT*16;
  int wave  = blockIdx.x*(blockDim.x>>5) + (threadIdx.x>>5);
  int mbase = wave<<4;
  if (mbase >= M) return;                 // wave-uniform exit: EXEC all-1 for WMMA
  int lane = threadIdx.x & 31;
  int half = lane >> 4;
  int r    = lane & 15;
  int row  = mbase + r;
  if (row >= M) row = M-1;                // clamp (stores guarded below)
  const float* Ap = A + (size_t)row*lda + (half<<1);
  v8f acc[NCT] = {};
  for (int k = 0; k < K; k += 4) {
    v2f a = *(const v2f*)(Ap + k);
    const float* Wk = W + (size_t)(k + (half<<1))*ldw + r;
#pragma unroll
    for (int ct = 0; ct < NCT; ++ct) {
      v2f b;
      b.x = Wk[ct*16];
      b.y = Wk[ct*16 + ldw];
      acc[ct] = __builtin_amdgcn_wmma_f32_16x16x4_f32(
          false, a, false, b, (short)0, acc[ct], false, false);
    }
  }
#pragma unroll
  for (int ct = 0; ct < NCT; ++ct) {
    int col = ct*16 + r;
    float bv = bias ? bias[col] : 0.f;
#pragma unroll
    for (int v = 0; v < 8; ++v) {
      int mr = mbase + v + (half<<3);     // D layout: M = v (+8 for upper half)
      if (mr < M) {
        float val = acc[ct][v] + bv;
        if (RELU) val = fmaxf(val, 0.f);
        float* cp = C + (size_t)mr*ldw + col;
        if (ADD) *cp += val; else *cp = val;
      }
    }
  }
}

// ---------------------------------------------------------------------------
// Small helper kernels
// ---------------------------------------------------------------------------
__global__ void fillk(float* p, float v, int n){
  int i = blockIdx.x*blockDim.x + threadIdx.x;
  if (i < n) p[i] = v;
}

__global__ void pad_wr1(const float* __restrict__ Wr1, float* __restrict__ Wr1p){
  int i = blockIdx.x*blockDim.x + threadIdx.x;
  if (i >= 3*RADP*64) return;
  int j = i & 63;
  int k = (i >> 6) % RADP;
  int l = i / (64*RADP);
  Wr1p[i] = (k < 130) ? Wr1[(l*130 + k)*64 + j] : 0.f;
}

// h1 planes init: h1d[n,c] = x[n,d] * Win1[c]
__global__ void init_h1(const float* __restrict__ x, const float* __restrict__ Win1,
                        float* __restrict__ h1x, float* __restrict__ h1y,
                        float* __restrict__ h1z)
{
  int i = blockIdx.x*blockDim.x + threadIdx.x;
  if (i >= NQ*CQ) return;
  int n = i >> 5, c = i & 31;
  float w = Win1[c];
  h1x[i] = x[n*3+0]*w;
  h1y[i] = x[n*3+1]*w;
  h1z[i] = x[n*3+2]*w;
}

// Per-edge geometric features + ef row gather (wave per edge)
__global__ void edge_init(const float* __restrict__ pair, const float* __restrict__ bppm,
                          const float* __restrict__ x, const int* __restrict__ src,
                          const int* __restrict__ dst, float* __restrict__ ef,
                          float* __restrict__ rvec, int E)
{
  int wid = blockIdx.x*(blockDim.x>>5) + (threadIdx.x>>5);
  if (wid >= E) return;
  int lane = threadIdx.x & 31;
  int s = src[wid], d = dst[wid];
  float rx = x[d*3+0]-x[s*3+0];
  float ry = x[d*3+1]-x[s*3+1];
  float rz = x[d*3+2]-x[s*3+2];
  float r  = sqrtf(rx*rx + ry*ry + rz*rz + 1e-12f);
  float inv = 1.f/(r + 1e-8f);
  int b  = s >> 9;
  int si = s & 511;
  int di = d & 511;
  const float* pr = pair + (((((size_t)b)*LQ + si)*LQ + di) << 7);
  float* e = ef + (size_t)wid*RADP;
  for (int c = lane; c < PQ; c += 32) e[c] = pr[c];
  if (lane == 0) {
    e[128] = bppm[(((size_t)b)*LQ + si)*LQ + di];
    e[129] = r;
    e[130] = 0.f;
    e[131] = 0.f;
    rvec[wid*4+0] = rx*inv;
    rvec[wid*4+1] = ry*inv;
    rvec[wid*4+2] = rz*inv;
    rvec[wid*4+3] = r;
  }
}

// dot1[e,c] = sum_d h1[src,c,d] * rhat[e,d]   (wave per edge, lane = channel)
__global__ void edge_dot1(const float* __restrict__ h1x, const float* __restrict__ h1y,
                          const float* __restrict__ h1z, const int* __restrict__ src,
                          const float* __restrict__ rvec, float* __restrict__ dot1, int E)
{
  int wid = blockIdx.x*(blockDim.x>>5) + (threadIdx.x>>5);
  if (wid >= E) return;
  int c = threadIdx.x & 31;
  int s = src[wid];
  float rx = rvec[wid*4+0], ry = rvec[wid*4+1], rz = rvec[wid*4+2];
  dot1[(size_t)wid*CQ + c] = h1x[s*CQ+c]*rx + h1y[s*CQ+c]*ry + h1z[s*CQ+c]*rz;
}

// Edge messages m0/m1 + attention logits + segment-max (wave per edge)
__global__ void edge_msg(const float* __restrict__ rad, const float* __restrict__ t00,
                         const float* __restrict__ t10, const float* __restrict__ tdot,
                         const float* __restrict__ hm1x, const float* __restrict__ hm1y,
                         const float* __restrict__ hm1z, const float* __restrict__ qb,
                         const int* __restrict__ src, const int* __restrict__ dst,
                         const float* __restrict__ rvec,
                         float* __restrict__ m0, float* __restrict__ m1x,
                         float* __restrict__ m1y, float* __restrict__ m1z,
                         float* __restrict__ logit, float* __restrict__ mx, int E)
{
  int wid = blockIdx.x*(blockDim.x>>5) + (threadIdx.x>>5);
  if (wid >= E) return;
  int c = threadIdx.x & 31;
  int s = src[wid], d = dst[wid];
  const float* rd = rad + (size_t)wid*(4*CQ);
  float w00 = rd[c], w01 = rd[CQ+c], w10 = rd[2*CQ+c], w11 = rd[3*CQ+c];
  float m0v = w00*t00[s*CQ+c] + w01*tdot[(size_t)wid*CQ+c];
  m0[(size_t)wid*CQ+c] = m0v;
  float rx = rvec[wid*4+0], ry = rvec[wid*4+1], rz = rvec[wid*4+2];
  float g = w10 * t10[s*CQ+c];
  m1x[(size_t)wid*CQ+c] = w11*hm1x[s*CQ+c] + g*rx;
  m1y[(size_t)wid*CQ+c] = w11*hm1y[s*CQ+c] + g*ry;
  m1z[(size_t)wid*CQ+c] = w11*hm1z[s*CQ+c] + g*rz;
  // logit[e,h] = (q[dst,h,:] . m0[e,h,:]) / sqrt(8), heads of 8 lanes
  float p = qb[d*CQ+c]*m0v;
  p = wred8(p);
  if ((c & 7) == 0) {
    float lg = p * 0.35355339059327373f;  // 1/sqrt(8)
    int h = c >> 3;
    logit[(size_t)wid*4 + h] = lg;
    atomicMaxFloat(mx + d*4 + h, lg);
  }
}

// a = exp(logit - mx[dst]); den += a  (thread per (edge, head))
__global__ void edge_exp(const float* __restrict__ mxv, const int* __restrict__ dst,
                         float* __restrict__ logit, float* __restrict__ den, int E)
{
  int i = blockIdx.x*blockDim.x + threadIdx.x;
  if (i >= E*4) return;
  int e = i >> 2, h = i & 3;
  int d = dst[e];
  float a = __expf(logit[i] - mxv[d*4+h]);
  logit[i] = a;
  atomicAdd(den + d*4 + h, a);
}

// alpha-weighted scatter of m0 / m1 planes into node aggregates
__global__ void edge_scatter(const float* __restrict__ abuf, const float* __restrict__ den,
                             const int* __restrict__ dst, const float* __restrict__ m0,
                             const float* __restrict__ m1x, const float* __restrict__ m1y,
                             const float* __restrict__ m1z, float* __restrict__ agg0,
                             float* __restrict__ agg1x, float* __restrict__ agg1y,
                             float* __restrict__ agg1z, int E)
{
  int wid = blockIdx.x*(blockDim.x>>5) + (threadIdx.x>>5);
  if (wid >= E) return;
  int c = threadIdx.x & 31;
  int d = dst[wid];
  int h = c >> 3;
  float alpha = abuf[(size_t)wid*4 + h] / (den[d*4+h] + 1e-9f);
  atomicAdd(agg0  + d*CQ + c, alpha * m0 [(size_t)wid*CQ+c]);
  atomicAdd(agg1x + d*CQ + c, alpha * m1x[(size_t)wid*CQ+c]);
  atomicAdd(agg1y + d*CQ + c, alpha * m1y[(size_t)wid*CQ+c]);
  atomicAdd(agg1z + d*CQ + c, alpha * m1z[(size_t)wid*CQ+c]);
}

// LayerNorm on h0, equivariant norm-gating on h1 (wave per node, lane = channel)
__global__ void node_norm(float* __restrict__ h0, float* __restrict__ h1x,
                          float* __restrict__ h1y, float* __restrict__ h1z,
                          const float* __restrict__ gamma0, const float* __restrict__ beta0,
                          const float* __restrict__ gamma1)
{
  int n = blockIdx.x*(blockDim.x>>5) + (threadIdx.x>>5);
  if (n >= NQ) return;
  int c = threadIdx.x & 31;
  float v  = h0[n*CQ+c];
  float mu = wred32(v) * (1.f/CQ);
  float dv = v - mu;
  float var = wred32(dv*dv) * (1.f/CQ);
  h0[n*CQ+c] = dv * rsqrtf(var + 1e-5f) * gamma0[c] + beta0[c];
  float ax = h1x[n*CQ+c], ay = h1y[n*CQ+c], az = h1z[n*CQ+c];
  float nrm = sqrtf(ax*ax + ay*ay + az*az + 1e-12f);
  float nmu = wred32(nrm) * (1.f/CQ);
  float nd  = nrm - nmu;
  float nv  = wred32(nd*nd) * (1.f/CQ);
  float nln = nd * rsqrtf(nv + 1e-5f) * gamma1[c];
  float sc  = nln / (nrm + 1e-8f);
  h1x[n*CQ+c] = ax*sc;
  h1y[n*CQ+c] = ay*sc;
  h1z[n*CQ+c] = az*sc;
}

// out[n,d] = x[n,d] + sum_c h1d[n,c] * Wout[c]
__global__ void out_k(const float* __restrict__ h1x, const float* __restrict__ h1y,
                      const float* __restrict__ h1z, const float* __restrict__ Wout,
                      const float* __restrict__ x, float* __restrict__ out)
{
  int n = blockIdx.x*(blockDim.x>>5) + (threadIdx.x>>5);
  if (n >= NQ) return;
  int c = threadIdx.x & 31;
  float w = Wout[c];
  float sx = wred32(h1x[n*CQ+c]*w);
  float sy = wred32(h1y[n*CQ+c]*w);
  float sz = wred32(h1z[n*CQ+c]*w);
  if (c == 0) {
    out[n*3+0] = x[n*3+0] + sx;
    out[n*3+1] = x[n*3+1] + sy;
    out[n*3+2] = x[n*3+2] + sz;
  }
}

// ---------------------------------------------------------------------------
// Host side
// ---------------------------------------------------------------------------
template<int NCT, bool RELU, bool ADD>
static void gemm(const float* A, const float* W, const float* b, float* C,
                 int M, int K, int lda, hipStream_t st)
{
  int waves  = (M + 15) / 16;
  int blocks = (waves + 3) / 4;
  wmma_gemm<NCT,RELU,ADD><<<blocks, 128, 0, st>>>(A, W, b, C, M, K, lda);
}

extern "C" void kernel_launch(void* const* d_in, const int* in_sizes, int n_in,
                              void* d_out, int out_size, void* d_ws, size_t ws_size,
                              hipStream_t stream)
{
  const float* seq   = (const float*)d_in[0];
  const float* pair  = (const float*)d_in[1];
  const float* bppm  = (const float*)d_in[2];
  const float* x0    = (const float*)d_in[3];
  const int*   esrc  = (const int*)  d_in[4];
  const int*   edst  = (const int*)  d_in[5];
  const float* Win0  = (const float*)d_in[6];
  const float* Win1  = (const float*)d_in[7];
  const float* Wr1   = (const float*)d_in[8];
  const float* br1   = (const float*)d_in[9];
  const float* Wr2   = (const float*)d_in[10];
  const float* br2   = (const float*)d_in[11];
  const float* Wm00  = (const float*)d_in[12];
  const float* Wm01  = (const float*)d_in[13];
  const float* Wm10  = (const float*)d_in[14];
  const float* Wm11  = (const float*)d_in[15];
  const float* Wq    = (const float*)d_in[16];
  const float* Wself0= (const float*)d_in[17];
  const float* Wself1= (const float*)d_in[18];
  const float* gamma0= (const float*)d_in[19];
  const float* beta0 = (const float*)d_in[20];
  const float* gamma1= (const float*)d_in[21];
  const float* Wout  = (const float*)d_in[22];
  float* out = (float*)d_out;

  const int E = in_sizes[4];

  // -------- workspace partition (all regions fully rewritten every call) ----
  float* wp = (float*)d_ws;
  auto alloc = [&](size_t nf)->float*{
    float* r = wp;
    wp += (nf + 63) & ~((size_t)63);
    return r;
  };
  float* h0    = alloc((size_t)NQ*CQ);
  float* h1x   = alloc((size_t)NQ*CQ);
  float* h1y   = alloc((size_t)NQ*CQ);
  float* h1z   = alloc((size_t)NQ*CQ);
  float* t00   = alloc((size_t)NQ*CQ);
  float* t10   = alloc((size_t)NQ*CQ);
  float* qb    = alloc((size_t)NQ*CQ);
  float* hm1x  = alloc((size_t)NQ*CQ);
  float* hm1y  = alloc((size_t)NQ*CQ);
  float* hm1z  = alloc((size_t)NQ*CQ);
  float* mx    = alloc((size_t)NQ*4);
  float* zreg  = alloc((size_t)NQ*(4 + 4*CQ));   // den + agg0 + agg1{x,y,z}
  float* den   = zreg;
  float* agg0  = zreg + NQ*4;
  float* agg1x = agg0 + NQ*CQ;
  float* agg1y = agg1x + NQ*CQ;
  float* agg1z = agg1y + NQ*CQ;
  float* Wr1p  = alloc((size_t)3*RADP*64);
  float* ef    = alloc((size_t)E*RADP);
  float* rvec  = alloc((size_t)E*4);
  float* rhid  = alloc((size_t)E*64);            // reused: dot1 / tdot after rad GEMM
  float* dot1  = rhid;
  float* tdot  = rhid + (size_t)E*CQ;
  float* rad   = alloc((size_t)E*4*CQ);
  float* m0b   = alloc((size_t)E*CQ);
  float* m1x   = alloc((size_t)E*CQ);
  float* m1y   = alloc((size_t)E*CQ);
  float* m1z   = alloc((size_t)E*CQ);
  float* logit = alloc((size_t)E*4);

  const int TB = 256;
  int eWaveBlocks = (E + 7) / 8;                  // 8 waves per 256-thread block
  int nWaveBlocks = (NQ + 7) / 8;

  // -------- setup ----------------------------------------------------------
  pad_wr1<<<(3*RADP*64 + TB-1)/TB, TB, 0, stream>>>(Wr1, Wr1p);
  gemm<2,false,false>(seq, Win0, nullptr, h0, NQ, D0Q, D0Q, stream);       // h0 = seq @ Win0
  init_h1<<<(NQ*CQ + TB-1)/TB, TB, 0, stream>>>(x0, Win1, h1x, h1y, h1z);
  edge_init<<<eWaveBlocks, TB, 0, stream>>>(pair, bppm, x0, esrc, edst, ef, rvec, E);

  // -------- layers ---------------------------------------------------------
  for (int l = 0; l < 3; ++l) {
    const float* wr1p = Wr1p + (size_t)l*RADP*64;
    const float* b1   = br1  + l*64;
    const float* wr2  = Wr2  + (size_t)l*64*128;
    const float* b2   = br2  + l*128;
    const float* wm00 = Wm00 + l*CQ*CQ;
    const float* wm01 = Wm01 + l*CQ*CQ;
    const float* wm10 = Wm10 + l*CQ*CQ;
    const float* wm11 = Wm11 + l*CQ*CQ;
    const float* wq   = Wq   + l*CQ*CQ;
    const float* ws0  = Wself0 + l*CQ*CQ;
    const float* ws1  = Wself1 + l*CQ*CQ;

    fillk<<<(NQ*4 + TB-1)/TB, TB, 0, stream>>>(mx, -INFINITY, NQ*4);
    fillk<<<(NQ*(4+4*CQ) + TB-1)/TB, TB, 0, stream>>>(zreg, 0.f, NQ*(4+4*CQ));

    // radial MLP (dominant FLOPs, fp32 WMMA)
    gemm<4,true ,false>(ef,   wr1p, b1, rhid, E, RADP, RADP, stream);      // [E,64]
    gemm<8,false,false>(rhid, wr2,  b2, rad,  E, 64,   64,   stream);      // [E,128]

    // node-level linear transforms (lifted out of edge loop)
    gemm<2,false,false>(h0,  wm00, nullptr, t00,  NQ, CQ, CQ, stream);
    gemm<2,false,false>(h0,  wm10, nullptr, t10,  NQ, CQ, CQ, stream);
    gemm<2,false,false>(h0,  wq,   nullptr, qb,   NQ, CQ, CQ, stream);
    gemm<2,false,false>(h1x, wm11, nullptr, hm1x, NQ, CQ, CQ, stream);
    gemm<2,false,false>(h1y, wm11, nullptr, hm1y, NQ, CQ, CQ, stream);
    gemm<2,false,false>(h1z, wm11, nullptr, hm1z, NQ, CQ, CQ, stream);

    // deg1 -> deg0 path
    edge_dot1<<<eWaveBlocks, TB, 0, stream>>>(h1x, h1y, h1z, esrc, rvec, dot1, E);
    gemm<2,false,false>(dot1, wm01, nullptr, tdot, E, CQ, CQ, stream);

    // messages + attention
    edge_msg<<<eWaveBlocks, TB, 0, stream>>>(rad, t00, t10, tdot, hm1x, hm1y, hm1z,
                                             qb, esrc, edst, rvec,
                                             m0b, m1x, m1y, m1z, logit, mx, E);
    edge_exp<<<(E*4 + TB-1)/TB, TB, 0, stream>>>(mx, edst, logit, den, E);
    edge_scatter<<<eWaveBlocks, TB, 0, stream>>>(logit, den, edst, m0b, m1x, m1y, m1z,
                                                 agg0, agg1x, agg1y, agg1z, E);

    // residual updates (accumulate variant of the WMMA GEMM)
    gemm<2,false,true>(agg0,  ws0, nullptr, h0,  NQ, CQ, CQ, stream);
    gemm<2,false,true>(agg1x, ws1, nullptr, h1x, NQ, CQ, CQ, stream);
    gemm<2,false,true>(agg1y, ws1, nullptr, h1y, NQ, CQ, CQ, stream);
    gemm<2,false,true>(agg1z, ws1, nullptr, h1z, NQ, CQ, CQ, stream);

    node_norm<<<nWaveBlocks, TB, 0, stream>>>(h0, h1x, h1y, h1z,
                                              gamma0 + l*CQ, beta0 + l*CQ, gamma1 + l*CQ);
  }

  // -------- output ---------------------------------------------------------
  out_k<<<nWaveBlocks, TB, 0, stream>>>(h1x, h1y, h1z, Wout, x0, out);
}